// LightGCL_40630390620838
// MI455X (gfx1250) — compile-verified
//
#include <hip/hip_runtime.h>
#include <hip/hip_bf16.h>

typedef __attribute__((ext_vector_type(2))) float v2f;
typedef __attribute__((ext_vector_type(8))) float v8f;
typedef __attribute__((__vector_size__(16))) int vi4;   // b128 payload type for async LDS builtins

#define N_U 100000
#define N_I 80000
#define NT  180000            // N_U + N_I
#define D   64
#define H   128
#define LAY 2
#define NE  3200000
#define BB  4096
#define BD  (BB * D)          // 262144

// ---------------------------------------------------------------------------
// gfx1250 async global->LDS copy (ASYNCcnt path), with safe fallback
// ---------------------------------------------------------------------------
#if defined(__gfx1250__) && __has_builtin(__builtin_amdgcn_global_load_async_to_lds_b128) && __has_builtin(__builtin_amdgcn_s_wait_asynccnt)
#define HAS_ASYNC_LDS 1
#else
#define HAS_ASYNC_LDS 0
#endif

__device__ __forceinline__ void lds_copy_b128(const float* gsrc, float* lds_dst) {
#if HAS_ASYNC_LDS
    __builtin_amdgcn_global_load_async_to_lds_b128(
        (__attribute__((address_space(1))) vi4*)gsrc,
        (__attribute__((address_space(3))) vi4*)lds_dst,
        0, 0);
#else
    float4 v = *(const float4*)gsrc;
    *(float4*)lds_dst = v;
#endif
}

__device__ __forceinline__ void lds_copy_wait() {
#if HAS_ASYNC_LDS
    __builtin_amdgcn_s_wait_asynccnt(0);
#endif
}

// ---------------------------------------------------------------------------
// concat x = [u_embeds ; i_embeds]
// ---------------------------------------------------------------------------
__global__ void concat_kernel(const float* __restrict__ ue,
                              const float* __restrict__ ie,
                              float* __restrict__ x) {
    int i = blockIdx.x * 256 + threadIdx.x;
    if (i < N_U * D) x[i] = ue[i];
    else if (i < NT * D) x[i] = ie[i - N_U * D];
}

// ---------------------------------------------------------------------------
// uu = adj @ i_hyper, ii = adj.T @ u_hyper  (32 lanes per edge, float4 each)
// ---------------------------------------------------------------------------
__global__ void spmm_hyper_kernel(const int* __restrict__ rows,
                                  const int* __restrict__ cols,
                                  const float* __restrict__ vals,
                                  const float* __restrict__ uhg,
                                  const float* __restrict__ ihg,
                                  float* __restrict__ uu,
                                  float* __restrict__ ii) {
    unsigned t = blockIdx.x * 256u + threadIdx.x;
    unsigned e = t >> 5;
    if (e >= NE) return;
    int h4 = (int)(t & 31u) * 4;
    int r = rows[e], c = cols[e];
    float v = vals[e];
    float4 iv = *(const float4*)(ihg + (size_t)c * H + h4);
    atomicAdd(&uu[(size_t)r * H + h4 + 0], v * iv.x);
    atomicAdd(&uu[(size_t)r * H + h4 + 1], v * iv.y);
    atomicAdd(&uu[(size_t)r * H + h4 + 2], v * iv.z);
    atomicAdd(&uu[(size_t)r * H + h4 + 3], v * iv.w);
    float4 uv = *(const float4*)(uhg + (size_t)r * H + h4);
    atomicAdd(&ii[(size_t)c * H + h4 + 0], v * uv.x);
    atomicAdd(&ii[(size_t)c * H + h4 + 1], v * uv.y);
    atomicAdd(&ii[(size_t)c * H + h4 + 2], v * uv.z);
    atomicAdd(&ii[(size_t)c * H + h4 + 3], v * uv.w);
}

// ---------------------------------------------------------------------------
// GCN bipartite spmm with adj_norm (16 lanes per edge, D=64)
// ---------------------------------------------------------------------------
__global__ void gcn_spmm_kernel(const int* __restrict__ rows,
                                const int* __restrict__ cols,
                                const float* __restrict__ vals,
                                const float* __restrict__ bn,
                                float* __restrict__ e1) {
    unsigned t = blockIdx.x * 256u + threadIdx.x;
    unsigned e = t >> 4;
    if (e >= NE) return;
    int c4 = (int)(t & 15u) * 4;
    int r = rows[e], c = cols[e];
    float v = vals[e];
    float4 bi = *(const float4*)(bn + (size_t)(N_U + c) * D + c4);
    atomicAdd(&e1[(size_t)r * D + c4 + 0], v * bi.x);
    atomicAdd(&e1[(size_t)r * D + c4 + 1], v * bi.y);
    atomicAdd(&e1[(size_t)r * D + c4 + 2], v * bi.z);
    atomicAdd(&e1[(size_t)r * D + c4 + 3], v * bi.w);
    float4 bu = *(const float4*)(bn + (size_t)r * D + c4);
    atomicAdd(&e1[(size_t)(N_U + c) * D + c4 + 0], v * bu.x);
    atomicAdd(&e1[(size_t)(N_U + c) * D + c4 + 1], v * bu.y);
    atomicAdd(&e1[(size_t)(N_U + c) * D + c4 + 2], v * bu.z);
    atomicAdd(&e1[(size_t)(N_U + c) * D + c4 + 3], v * bu.w);
}

// ---------------------------------------------------------------------------
// column sums / sum-of-squares for BatchNorm
// ---------------------------------------------------------------------------
__global__ __launch_bounds__(256) void colstats_kernel(const float* __restrict__ x,
                                                       float* __restrict__ cs) {
    __shared__ float ls[256], lss[256];
    int col = threadIdx.x & 63;
    int grp = threadIdx.x >> 6;
    float s = 0.f, ss = 0.f;
    for (int r = blockIdx.x * 4 + grp; r < NT; r += gridDim.x * 4) {
        float v = x[(size_t)r * D + col];
        s += v; ss += v * v;
    }
    ls[threadIdx.x] = s; lss[threadIdx.x] = ss;
    __syncthreads();
    if (threadIdx.x < 64) {
        float a  = ls[threadIdx.x]  + ls[threadIdx.x + 64]  + ls[threadIdx.x + 128]  + ls[threadIdx.x + 192];
        float b2 = lss[threadIdx.x] + lss[threadIdx.x + 64] + lss[threadIdx.x + 128] + lss[threadIdx.x + 192];
        atomicAdd(&cs[col], a);
        atomicAdd(&cs[64 + col], b2);
    }
}

__global__ void bn_apply_kernel(const float* __restrict__ x,
                                float* __restrict__ bn,
                                const float* __restrict__ cs,
                                const float* __restrict__ w,
                                const float* __restrict__ b) {
    int i = blockIdx.x * 256 + threadIdx.x;
    if (i >= NT * D) return;
    int col = i & 63;
    const float invN = 1.0f / (float)NT;
    float mu  = cs[col] * invN;
    float var = cs[64 + col] * invN - mu * mu;
    bn[i] = (x[i] - mu) * rsqrtf(var + 1e-5f) * w[col] + b[col];
}

// ---------------------------------------------------------------------------
// mid = U^T @ X  -> [128,64], split-K with register tile + f32 atomics
// ---------------------------------------------------------------------------
__global__ __launch_bounds__(256) void atb_kernel(const float* __restrict__ U,   // [K,128]
                                                  const float* __restrict__ X,   // [K,64]
                                                  float* __restrict__ mid,       // [128,64]
                                                  int K) {
    int n  = threadIdx.x & 63;
    int mg = threadIdx.x >> 6;     // 0..3
    int m0 = mg * 32;
    float acc[32];
#pragma unroll
    for (int mm = 0; mm < 32; ++mm) acc[mm] = 0.f;
    int chunk = (K + gridDim.x - 1) / gridDim.x;
    int k0 = blockIdx.x * chunk;
    int k1 = min(K, k0 + chunk);
    for (int k = k0; k < k1; ++k) {
        float xv = X[(size_t)k * D + n];
        const float* urow = U + (size_t)k * H + m0;
#pragma unroll
        for (int mm = 0; mm < 32; ++mm) acc[mm] += urow[mm] * xv;
    }
#pragma unroll
    for (int mm = 0; mm < 32; ++mm)
        atomicAdd(&mid[(m0 + mm) * D + n], acc[mm]);
}

// ---------------------------------------------------------------------------
// e2 = U @ mid : [nrows,128] x [128,64], f32 WMMA 16x16x4.
// 8 waves/block, one 16-row stripe per wave, 4 N-tiles, K=128 (32 k-steps).
// mid staged via async global->LDS.
// ---------------------------------------------------------------------------
__global__ __launch_bounds__(256) void hyp2_wmma_kernel(const float* __restrict__ U,
                                                        const float* __restrict__ mid,
                                                        float* __restrict__ out,
                                                        int nrows) {
    __shared__ float smid[H * D];   // 32 KB
    for (int i = threadIdx.x * 4; i < H * D; i += 256 * 4)
        lds_copy_b128(mid + i, &smid[i]);
    lds_copy_wait();
    __syncthreads();

    int wave = threadIdx.x >> 5;
    int lane = threadIdx.x & 31;
    int r0 = (blockIdx.x * 8 + wave) * 16;
    if (r0 >= nrows) return;                 // wave-uniform exit

    int m    = lane & 15;
    int rowc = min(r0 + m, nrows - 1);       // clamp so EXEC stays all-ones
    v8f acc0 = {}, acc1 = {}, acc2 = {}, acc3 = {};

    for (int kt = 0; kt < 32; ++kt) {
        int kb = kt * 4 + ((lane < 16) ? 0 : 2);
        v2f a; a.x = U[(size_t)rowc * H + kb]; a.y = U[(size_t)rowc * H + kb + 1];
        int n = lane & 15;
        v2f b0; b0.x = smid[kb * D + n];        b0.y = smid[(kb + 1) * D + n];
        v2f b1; b1.x = smid[kb * D + 16 + n];   b1.y = smid[(kb + 1) * D + 16 + n];
        v2f b2; b2.x = smid[kb * D + 32 + n];   b2.y = smid[(kb + 1) * D + 32 + n];
        v2f b3; b3.x = smid[kb * D + 48 + n];   b3.y = smid[(kb + 1) * D + 48 + n];
        acc0 = __builtin_amdgcn_wmma_f32_16x16x4_f32(false, a, false, b0, (short)0, acc0, false, false);
        acc1 = __builtin_amdgcn_wmma_f32_16x16x4_f32(false, a, false, b1, (short)0, acc1, false, false);
        acc2 = __builtin_amdgcn_wmma_f32_16x16x4_f32(false, a, false, b2, (short)0, acc2, false, false);
        acc3 = __builtin_amdgcn_wmma_f32_16x16x4_f32(false, a, false, b3, (short)0, acc3, false, false);
    }
#pragma unroll
    for (int r = 0; r < 8; ++r) {
        int mr = r + ((lane < 16) ? 0 : 8);
        int grow = r0 + mr;
        if (grow < nrows) {
            float* o = out + (size_t)grow * D + (lane & 15);
            o[0]  = acc0[r];
            o[16] = acc1[r];
            o[32] = acc2[r];
            o[48] = acc3[r];
        }
    }
}

// ---------------------------------------------------------------------------
// gather per-layer loss slices: e1u, e1i, e2u, e2i, e3u=e1u+xu, e3i=e1i+xi
// ---------------------------------------------------------------------------
__global__ void slice_kernel(const float* __restrict__ e1,
                             const float* __restrict__ e2,
                             const float* __restrict__ x,
                             const int* __restrict__ uids,
                             const int* __restrict__ pos,
                             float* __restrict__ sl) {
    int i = blockIdx.x * 256 + threadIdx.x;
    if (i >= BD) return;
    int s = i >> 6, c = i & 63;
    size_t u = (size_t)uids[s] * D + c;
    size_t p = (size_t)(N_U + pos[s]) * D + c;
    float e1u = e1[u], e1i = e1[p];
    sl[0 * BD + i] = e1u;
    sl[1 * BD + i] = e1i;
    sl[2 * BD + i] = e2[u];
    sl[3 * BD + i] = e2[p];
    sl[4 * BD + i] = e1u + x[u];
    sl[5 * BD + i] = e1i + x[p];
}

// ---------------------------------------------------------------------------
// row-wise l2 normalization in place (one wave per row of 64)
// ---------------------------------------------------------------------------
__global__ __launch_bounds__(256) void normalize_rows_kernel(float* __restrict__ p,
                                                             int nrows) {
    int row = blockIdx.x * 8 + (threadIdx.x >> 5);
    if (row >= nrows) return;
    int lane = threadIdx.x & 31;
    size_t base = (size_t)row * D + lane * 2;
    float2 v = *(const float2*)(p + base);
    float ss = v.x * v.x + v.y * v.y;
    ss += __shfl_xor(ss, 1);  ss += __shfl_xor(ss, 2);
    ss += __shfl_xor(ss, 4);  ss += __shfl_xor(ss, 8);
    ss += __shfl_xor(ss, 16);
    float inv = 1.0f / fmaxf(sqrtf(ss), 1e-12f);
    v.x *= inv; v.y *= inv;
    *(float2*)(p + base) = v;
}

// ---------------------------------------------------------------------------
// x = x + e1 + ALPHA * l2norm_row(e2)     (one wave per row)
// ---------------------------------------------------------------------------
__global__ __launch_bounds__(256) void update_x_kernel(float* __restrict__ x,
                                                       const float* __restrict__ e1,
                                                       const float* __restrict__ e2) {
    int row = blockIdx.x * 8 + (threadIdx.x >> 5);
    if (row >= NT) return;
    int lane = threadIdx.x & 31;
    size_t base = (size_t)row * D + lane * 2;
    float2 ev = *(const float2*)(e2 + base);
    float ss = ev.x * ev.x + ev.y * ev.y;
    ss += __shfl_xor(ss, 1);  ss += __shfl_xor(ss, 2);
    ss += __shfl_xor(ss, 4);  ss += __shfl_xor(ss, 8);
    ss += __shfl_xor(ss, 16);
    float inv = 0.2f / fmaxf(sqrtf(ss), 1e-12f);
    float2 xv  = *(const float2*)(x + base);
    float2 e1v = *(const float2*)(e1 + base);
    float2 o;
    o.x = xv.x + e1v.x + ev.x * inv;
    o.y = xv.y + e1v.y + ev.y * inv;
    *(float2*)(x + base) = o;
}

// ---------------------------------------------------------------------------
// fused contrast on PRE-NORMALIZED slices:
//   sum_i ( log(sum_j exp(<a_i,b_j>/T)) - <a_i,b_i>/T )
// 4 waves/block, each wave owns a 16-row stripe; B tile shared in LDS and
// fetched via async global->LDS by wave 0. WMMA f32 16x16x4, K=64.
// ---------------------------------------------------------------------------
__global__ __launch_bounds__(128) void contrast_kernel(const float* __restrict__ A,
                                                       const float* __restrict__ Bm,
                                                       float invTemp,
                                                       float* __restrict__ acc) {
    __shared__ float at[4][16 * D];   // 16 KB
    __shared__ float bt[16 * D];      // 4 KB
    int wave = threadIdx.x >> 5;
    int lane = threadIdx.x & 31;
    int rt = blockIdx.x * 4 + wave;   // row tile 0..255

    // stage this wave's (already normalized) A tile
    for (int i = lane * 4; i < 16 * D; i += 32 * 4)
        lds_copy_b128(A + (size_t)rt * 16 * D + i, &at[wave][i]);
    lds_copy_wait();
    __syncthreads();

    float pex[8];
#pragma unroll
    for (int r = 0; r < 8; ++r) pex[r] = 0.f;
    float dval = 0.f;
    int mn = lane & 15;

    for (int ct = 0; ct < BB / 16; ++ct) {
        __syncthreads();               // protect bt readers from next overwrite
        if (wave == 0) {
            for (int i = lane * 4; i < 16 * D; i += 32 * 4)
                lds_copy_b128(Bm + (size_t)ct * 16 * D + i, &bt[i]);
            lds_copy_wait();
        }
        __syncthreads();

        v8f cacc = {};
        for (int kt = 0; kt < 16; ++kt) {
            int kb = kt * 4 + ((lane < 16) ? 0 : 2);
            v2f a; a.x = at[wave][mn * D + kb]; a.y = at[wave][mn * D + kb + 1];
            v2f b; b.x = bt[mn * D + kb];       b.y = bt[mn * D + kb + 1];
            cacc = __builtin_amdgcn_wmma_f32_16x16x4_f32(false, a, false, b, (short)0, cacc, false, false);
        }
#pragma unroll
        for (int r = 0; r < 8; ++r) {
            float s = cacc[r] * invTemp;
            pex[r] += expf(s);
            int mtile = r + ((lane < 16) ? 0 : 8);
            if (ct == rt && mtile == mn) dval = s;   // diagonal element
        }
    }

    // row-wise reduction within each 16-lane half-wave
    float local = 0.f;
#pragma unroll
    for (int r = 0; r < 8; ++r) {
        float s = pex[r];
        s += __shfl_xor(s, 1, 16); s += __shfl_xor(s, 2, 16);
        s += __shfl_xor(s, 4, 16); s += __shfl_xor(s, 8, 16);
        int srcLane = (lane < 16) ? r : (24 + r);
        float d = __shfl(dval, srcLane, 32);
        if ((lane & 15) == 0) local += logf(s) - d;
    }
    if ((lane & 15) == 0) atomicAdd(acc, local);
}

// ---------------------------------------------------------------------------
// BPR loss (one wave per sample)
// ---------------------------------------------------------------------------
__global__ __launch_bounds__(256) void bpr_kernel(const float* __restrict__ x,
                                                  const int* __restrict__ uids,
                                                  const int* __restrict__ pos,
                                                  const int* __restrict__ neg,
                                                  float* __restrict__ acc) {
    int s = blockIdx.x * 8 + (threadIdx.x >> 5);
    if (s >= BB) return;
    int lane = threadIdx.x & 31;
    size_t ub = (size_t)uids[s] * D + lane * 2;
    size_t pb = (size_t)(N_U + pos[s]) * D + lane * 2;
    size_t nb = (size_t)(N_U + neg[s]) * D + lane * 2;
    float2 uv = *(const float2*)(x + ub);
    float2 pv = *(const float2*)(x + pb);
    float2 nv = *(const float2*)(x + nb);
    float d = uv.x * (pv.x - nv.x) + uv.y * (pv.y - nv.y);
    d += __shfl_xor(d, 1);  d += __shfl_xor(d, 2);
    d += __shfl_xor(d, 4);  d += __shfl_xor(d, 8);
    d += __shfl_xor(d, 16);
    if (lane == 0) {
        float diff = d * (1.0f / (float)D);
        float sg = 1.0f / (1.0f + expf(-diff));
        atomicAdd(acc, -logf(sg + 1e-15f));
    }
}

// ---------------------------------------------------------------------------
// sum of squares reduction
// ---------------------------------------------------------------------------
__global__ void sumsq_kernel(const float* __restrict__ p, long long n,
                             float* __restrict__ acc) {
    long long i = (long long)blockIdx.x * 256 + threadIdx.x;
    float s = 0.f;
    for (; i < n; i += (long long)gridDim.x * 256) { float v = p[i]; s += v * v; }
    s += __shfl_xor(s, 1);  s += __shfl_xor(s, 2);
    s += __shfl_xor(s, 4);  s += __shfl_xor(s, 8);
    s += __shfl_xor(s, 16);
    if ((threadIdx.x & 31) == 0) atomicAdd(acc, s);
}

__global__ void finalize_kernel(const float* __restrict__ scal,
                                float* __restrict__ out) {
    const float invB = 1.0f / (float)BB;
    float loss_r = scal[0] * invB;
    float ls  = scal[1] * invB * 0.1f;    // LAM1
    float ls2 = scal[2] * invB * 0.05f;   // LAM2
    float loss_s_total = (ls + ls2) / (float)LAY;
    float loss_reg = 1e-5f * scal[3];
    out[0] = loss_r + loss_s_total + loss_reg;
    out[1] = loss_r;
    out[2] = loss_s_total;
}

// ---------------------------------------------------------------------------
extern "C" void kernel_launch(void* const* d_in, const int* in_sizes, int n_in,
                              void* d_out, int out_size, void* d_ws, size_t ws_size,
                              hipStream_t stream) {
    const float* u_embeds = (const float*)d_in[0];
    const float* i_embeds = (const float*)d_in[1];
    const float* u_hyper  = (const float*)d_in[2];
    const float* i_hyper  = (const float*)d_in[3];
    const float* bn_w     = (const float*)d_in[4];
    const float* bn_b     = (const float*)d_in[5];
    const float* adj_norm = (const float*)d_in[6];
    const float* adj_vals = (const float*)d_in[7];
    const int*   adj_rows = (const int*)d_in[8];
    const int*   adj_cols = (const int*)d_in[9];
    const int*   uids     = (const int*)d_in[10];
    const int*   pos      = (const int*)d_in[12];
    const int*   neg      = (const int*)d_in[13];
    float* out = (float*)d_out;

    float* ws = (float*)d_ws;
    size_t o = 0;
    float* uu  = ws + o;  o += (size_t)N_U * H;     // 12.8 M
    float* ii  = ws + o;  o += (size_t)N_I * H;     // 10.24 M
    float* x   = ws + o;  o += (size_t)NT * D;      // 11.52 M
    float* bn  = ws + o;  o += (size_t)NT * D;
    float* e1  = ws + o;  o += (size_t)NT * D;
    float* e2  = ws + o;  o += (size_t)NT * D;
    float* mid_u = ws + o; o += H * D;              // 8192
    float* mid_i = ws + o; o += H * D;              // 8192
    float* cs    = ws + o; o += 2 * D;              // 128 (contiguous with mids)
    float* scal  = ws + o; o += 8;
    float* slices = ws + o; o += (size_t)2 * 6 * BD;
    (void)o; (void)ws_size; (void)in_sizes; (void)n_in; (void)out_size;

    // zero accumulated buffers (graph-capturable memset nodes)
    (void)hipMemsetAsync(uu, 0, ((size_t)N_U * H + (size_t)N_I * H) * sizeof(float), stream);
    (void)hipMemsetAsync(scal, 0, 8 * sizeof(float), stream);

    concat_kernel<<<(NT * D + 255) / 256, 256, 0, stream>>>(u_embeds, i_embeds, x);

    spmm_hyper_kernel<<<(NE * 32u + 255u) / 256u, 256, 0, stream>>>(
        adj_rows, adj_cols, adj_vals, u_hyper, i_hyper, uu, ii);

    for (int l = 0; l < LAY; ++l) {
        (void)hipMemsetAsync(mid_u, 0, (2 * H * D + 2 * D) * sizeof(float), stream); // mids + colstats
        colstats_kernel<<<1024, 256, 0, stream>>>(x, cs);
        bn_apply_kernel<<<(NT * D + 255) / 256, 256, 0, stream>>>(
            x, bn, cs, bn_w + l * D, bn_b + l * D);

        (void)hipMemsetAsync(e1, 0, (size_t)NT * D * sizeof(float), stream);
        gcn_spmm_kernel<<<(NE * 16u + 255u) / 256u, 256, 0, stream>>>(
            adj_rows, adj_cols, adj_norm, bn, e1);

        atb_kernel<<<128, 256, 0, stream>>>(uu, bn, mid_u, N_U);
        atb_kernel<<<128, 256, 0, stream>>>(ii, bn + (size_t)N_U * D, mid_i, N_I);

        hyp2_wmma_kernel<<<(N_U + 127) / 128, 256, 0, stream>>>(uu, mid_u, e2, N_U);
        hyp2_wmma_kernel<<<(N_I + 127) / 128, 256, 0, stream>>>(
            ii, mid_i, e2 + (size_t)N_U * D, N_I);

        slice_kernel<<<(BD + 255) / 256, 256, 0, stream>>>(
            e1, e2, x, uids, pos, slices + (size_t)l * 6 * BD);

        update_x_kernel<<<(NT + 7) / 8, 256, 0, stream>>>(x, e1, e2);
    }

    // BPR loss on final embeddings (still in x)
    bpr_kernel<<<(BB + 7) / 8, 256, 0, stream>>>(x, uids, pos, neg, &scal[0]);

    // normalize all 12 slices once (contrast consumes only l2-normalized rows)
    normalize_rows_kernel<<<(12 * BB + 7) / 8, 256, 0, stream>>>(slices, 12 * BB);

    // contrastive losses: invTemp = 1/0.2 = 5 for both T1 and T2
    for (int l = 0; l < LAY; ++l) {
        float* sl = slices + (size_t)l * 6 * BD;
        // loss_s:  (e3u vs e1u), (e3i vs e1i)
        contrast_kernel<<<BB / 64, 128, 0, stream>>>(sl + 4 * BD, sl + 0 * BD, 5.0f, &scal[1]);
        contrast_kernel<<<BB / 64, 128, 0, stream>>>(sl + 5 * BD, sl + 1 * BD, 5.0f, &scal[1]);
        // loss_s2: (e1u vs e2u), (e1i vs e2i)
        contrast_kernel<<<BB / 64, 128, 0, stream>>>(sl + 0 * BD, sl + 2 * BD, 5.0f, &scal[2]);
        contrast_kernel<<<BB / 64, 128, 0, stream>>>(sl + 1 * BD, sl + 3 * BD, 5.0f, &scal[2]);
    }

    // regularization sums of squares
    sumsq_kernel<<<2048, 256, 0, stream>>>(u_embeds, (long long)N_U * D, &scal[3]);
    sumsq_kernel<<<2048, 256, 0, stream>>>(i_embeds, (long long)N_I * D, &scal[3]);
    sumsq_kernel<<<2048, 256, 0, stream>>>(u_hyper,  (long long)N_U * H, &scal[3]);
    sumsq_kernel<<<2048, 256, 0, stream>>>(i_hyper,  (long long)N_I * H, &scal[3]);
    sumsq_kernel<<<1, 256, 0, stream>>>(bn_w, (long long)LAY * D, &scal[3]);
    sumsq_kernel<<<1, 256, 0, stream>>>(bn_b, (long long)LAY * D, &scal[3]);

    finalize_kernel<<<1, 1, 0, stream>>>(scal, out);
}